// ColumnwiseQuantizationOptim_20572893347947
// MI455X (gfx1250) — compile-verified
//
#include <hip/hip_runtime.h>
#include <stdint.h>

#define ROWS 4096
#define COLS 2048
#define NLEV 16
#define EPS  1e-6f

// exp(100*d) with d = -|u - q|  ->  exp2(-(100/ln2)*|u-q|)
#define NEG_T_OVER_LN2 (-144.269504088896f)
#define LN2F (0.69314718055994531f)

#define BLK        256   // threads per block; thread <-> one column
#define STAGE_ROWS 16    // rows per TDM tile
#define CHUNK_ROWS 128   // rows per blockIdx.y
#define NSTG       (CHUNK_ROWS / STAGE_ROWS)   // 8 pipeline stages

typedef float        v2f __attribute__((ext_vector_type(2)));
typedef float        v8f __attribute__((ext_vector_type(8)));
typedef unsigned int v4u __attribute__((ext_vector_type(4)));
typedef int          v4i __attribute__((ext_vector_type(4)));
typedef int          v8i __attribute__((ext_vector_type(8)));

#if __has_builtin(__builtin_amdgcn_tensor_load_to_lds) && __has_builtin(__builtin_amdgcn_s_wait_tensorcnt)
#define USE_TDM 1
#else
#define USE_TDM 0
#endif

#if __has_builtin(__builtin_amdgcn_wmma_f32_16x16x4_f32)
#define USE_WMMA 1
#else
#define USE_WMMA 0
#endif

#if USE_TDM
// One TDM descriptor: 2-D tile of STAGE_ROWS x BLK f32 elements out of a
// row-major [ROWS][COLS] tensor, landing contiguously in LDS.
// D# packing per CDNA5 ISA.
// 6-arg builtin form (clang-23 / therock-10.0 headers):
//   (uint32x4 g0, int32x8 g1, int32x4 g2, int32x4 g3, int32x8 g4, i32 cpol)
__device__ __forceinline__ void tdm_load_tile(uint32_t lds_byte_addr, const float* gsrc) {
  uint64_t ga = (uint64_t)(uintptr_t)gsrc;
  v4u g0;
  g0[0] = 1u;                                                 // count=1 (valid), user mode
  g0[1] = lds_byte_addr;                                      // lds_addr
  g0[2] = (uint32_t)ga;                                       // global_addr[31:0]
  g0[3] = ((uint32_t)(ga >> 32) & 0x01FFFFFFu) | (2u << 30);  // global_addr[56:32] | type=2
  v8i g1;
  g1[0] = (int)(2u << 16);            // workgroup_mask=0 (no cluster), data_size=2 -> 4 bytes
  g1[1] = (int)((uint32_t)BLK << 16); // tensor_dim0[15:0]=256 (tile-sized, never OOB)
  g1[2] = (int)((uint32_t)STAGE_ROWS << 16); // tensor_dim0 hi=0 ; tensor_dim1[15:0]=16
  g1[3] = (int)((uint32_t)BLK << 16); // tensor_dim1 hi=0 ; tile_dim0=256
  g1[4] = (int)STAGE_ROWS;            // tile_dim1=16 ; tile_dim2=0
  g1[5] = (int)COLS;                  // tensor_dim0_stride[31:0] = 2048 elements
  g1[6] = 0;                          // stride hi ; tensor_dim1_stride lo
  g1[7] = 0;
  v4i z4 = {0, 0, 0, 0};              // groups 2/3 unused (2-D tensor)
  v8i z8 = {0, 0, 0, 0, 0, 0, 0, 0};
  __builtin_amdgcn_tensor_load_to_lds(g0, g1, z4, z4, z8, 0);
}
#endif

__global__ void zero_ws_kernel(float* __restrict__ p, int n) {
  int i = blockIdx.x * blockDim.x + threadIdx.x;
  if (i < n) p[i] = 0.0f;
}

// Main kernel: grid (COLS/BLK, ROWS/CHUNK_ROWS) = (8, 32), 256 threads.
// Thread owns one column; loops CHUNK_ROWS rows fed by TDM-double-buffered LDS tiles.
__global__ __launch_bounds__(BLK) void softquant_main(
    const float* __restrict__ W, const float* __restrict__ wmin,
    const float* __restrict__ wmax, float* __restrict__ out,
    float* __restrict__ binmass) {
  __shared__ float tiles[2][STAGE_ROWS * BLK];   // 2 x 16 KB

  const int tid   = threadIdx.x;
  const int cbase = blockIdx.x * BLK;
  const int c     = cbase + tid;
  const int r0    = blockIdx.y * CHUNK_ROWS;
  const int wave  = tid >> 5;

  // Per-column clamp params (cheap, recomputed per block).
  float mn  = wmin[c];
  float mx  = wmax[c];
  float mnc = fminf(mn, mx - EPS);
  float mxc = fmaxf(mx, mnc + EPS);
  float rng = mxc - mnc;
  float inv_rng = 1.0f / (rng + EPS);

  float bins[NLEV];
#pragma unroll
  for (int l = 0; l < NLEV; ++l) bins[l] = 0.0f;

  auto stage_src = [&](int s) {
    return W + (size_t)(r0 + s * STAGE_ROWS) * COLS + cbase;
  };

#if USE_TDM
  if (wave == 0) tdm_load_tile((uint32_t)(uintptr_t)&tiles[0][0], stage_src(0));
#else
  {
    const float* g = stage_src(0);
    for (int i = tid; i < STAGE_ROWS * BLK; i += BLK)
      tiles[0][i] = g[(size_t)(i >> 8) * COLS + (i & (BLK - 1))];
  }
#endif

  for (int s = 0; s < NSTG; ++s) {
    const int nb = (s + 1) & 1;
#if USE_TDM
    if (wave == 0) {
      if (s + 1 < NSTG) {
        tdm_load_tile((uint32_t)(uintptr_t)&tiles[nb][0], stage_src(s + 1));
        __builtin_amdgcn_s_wait_tensorcnt(1);   // stage s complete (in-order)
      } else {
        __builtin_amdgcn_s_wait_tensorcnt(0);
      }
    }
#else
    if (s + 1 < NSTG) {
      const float* g = stage_src(s + 1);
      for (int i = tid; i < STAGE_ROWS * BLK; i += BLK)
        tiles[nb][i] = g[(size_t)(i >> 8) * COLS + (i & (BLK - 1))];
    }
#endif
    __syncthreads();   // stage s data visible to all waves

    const float* buf = tiles[s & 1];
    for (int r = 0; r < STAGE_ROWS; ++r) {
      float w = buf[r * BLK + tid];
      float u = (w - mnc) * inv_rng;          // in (0,1)
      float e[NLEV];
      float Z = 0.0f, S = 0.0f;
#pragma unroll
      for (int l = 0; l < NLEV; ++l) {
        float q  = (float)l * (1.0f / 15.0f);
        float t  = fabsf(u - q);
        float ev = __builtin_amdgcn_exp2f(t * NEG_T_OVER_LN2);  // v_exp_f32
        e[l] = ev;
        Z += ev;
        S = fmaf(ev, q, S);
      }
      float inv = 1.0f / Z;   // Z >= ~0.035, safe
      out[(size_t)(r0 + s * STAGE_ROWS + r) * COLS + c] = fmaf(S * inv, rng, mnc);
#pragma unroll
      for (int l = 0; l < NLEV; ++l) bins[l] = fmaf(e[l], inv, bins[l]);
    }
    __syncthreads();   // done reading buf before its TDM reuse next+1 iter
  }

#pragma unroll
  for (int l = 0; l < NLEV; ++l) atomicAdd(&binmass[c * NLEV + l], bins[l]);
}

// Entropy: per-column -sum p*log(p+eps) with p = mass/(sum+eps), then grand sum.
// Grand sum done as ones(16x4) x elp(4x16) f32 WMMA accumulation on wave 0:
// D[m][n] += sum_k elp[col k][level n]; all D rows identical, reduce D[0][0..15].
__global__ __launch_bounds__(256) void entropy_reduce(
    const float* __restrict__ binmass, float* __restrict__ out_scalar) {
  __shared__ float elp[512 * NLEV];   // 32 KB, one pass = 512 columns
  const int tid = threadIdx.x;

#if USE_WMMA
  v8f acc = {0.f, 0.f, 0.f, 0.f, 0.f, 0.f, 0.f, 0.f};
#else
  float acc_scalar = 0.0f;
#endif

  for (int pass = 0; pass < COLS / 512; ++pass) {
    for (int cl = tid; cl < 512; cl += 256) {
      int cidx = pass * 512 + cl;
      float m[NLEV];
      float ssum = 0.0f;
#pragma unroll
      for (int l = 0; l < NLEV; ++l) {
        m[l] = binmass[cidx * NLEV + l];
        ssum += m[l];
      }
      float invs = 1.0f / (ssum + EPS);
#pragma unroll
      for (int l = 0; l < NLEV; ++l) {
        float p  = m[l] * invs;
        float lg = LN2F * __builtin_amdgcn_logf(p + EPS);  // ln via v_log_f32
        elp[cl * NLEV + l] = -(p * lg);
      }
    }
    __syncthreads();

#if USE_WMMA
    if (tid < 32) {            // wave 0, EXEC all ones
      int n  = tid & 15;       // N = level
      int kb = (tid >> 4) * 2; // lanes 0-15 hold K=0,1 ; lanes 16-31 hold K=2,3
      for (int ch = 0; ch < 128; ++ch) {   // 4 columns per WMMA
        int col0 = ch * 4 + kb;
        v2f b;
        b.x = elp[(col0 + 0) * NLEV + n];
        b.y = elp[(col0 + 1) * NLEV + n];
        v2f a;
        a.x = 1.0f; a.y = 1.0f;            // all-ones A: layout-agnostic
        acc = __builtin_amdgcn_wmma_f32_16x16x4_f32(
            false, a, false, b, (short)0, acc, false, false);
      }
    }
#else
    if (tid < 32) {
      for (int i = tid; i < 512 * NLEV; i += 32) acc_scalar += elp[i];
    }
#endif
    __syncthreads();
  }

  if (tid < 32) {
#if USE_WMMA
    float v = acc[0];          // lanes 0-15: D[0][n] = per-level grand sums
#else
    float v = acc_scalar;
#endif
    v += __shfl_xor(v, 8);
    v += __shfl_xor(v, 4);
    v += __shfl_xor(v, 2);
    v += __shfl_xor(v, 1);
    if (tid == 0) out_scalar[0] = v;
  }
}

extern "C" void kernel_launch(void* const* d_in, const int* in_sizes, int n_in,
                              void* d_out, int out_size, void* d_ws, size_t ws_size,
                              hipStream_t stream) {
  (void)in_sizes; (void)n_in; (void)out_size; (void)ws_size;
  const float* W    = (const float*)d_in[0];
  const float* wmin = (const float*)d_in[1];
  const float* wmax = (const float*)d_in[2];
  float* out     = (float*)d_out;
  float* binmass = (float*)d_ws;          // COLS*NLEV f32 = 128 KB scratch

  const int nbm = COLS * NLEV;
  zero_ws_kernel<<<(nbm + 255) / 256, 256, 0, stream>>>(binmass, nbm);

  dim3 grid(COLS / BLK, ROWS / CHUNK_ROWS);   // (8, 32)
  softquant_main<<<grid, BLK, 0, stream>>>(W, wmin, wmax, out, binmass);

  entropy_reduce<<<1, 256, 0, stream>>>(binmass, out + (size_t)ROWS * COLS);
}